// GraphTransNet_86930138071446
// MI455X (gfx1250) — compile-verified
//
#include <hip/hip_runtime.h>
#include <math.h>

typedef __attribute__((ext_vector_type(2))) float v2f;
typedef __attribute__((ext_vector_type(8))) float v8f;

// ---------------- order-preserving float<->uint key for atomic max ----------
__device__ __forceinline__ unsigned fkey(float f) {
    unsigned b = __float_as_uint(f);
    return (b & 0x80000000u) ? ~b : (b | 0x80000000u);
}
__device__ __forceinline__ float funkey(unsigned u) {
    unsigned b = (u & 0x80000000u) ? (u & 0x7FFFFFFFu) : ~u;
    return __uint_as_float(b);
}
__device__ __forceinline__ bool finitef(float f) {
    return (__float_as_uint(f) & 0x7F800000u) != 0x7F800000u;
}

// ---------------- zero fill -------------------------------------------------
__global__ void fill_zero_k(float* __restrict__ p, int n) {
    int i = blockIdx.x * blockDim.x + threadIdx.x;
    if (i < n) p[i] = 0.0f;
}

// ---------------- fp32 WMMA GEMM: out[N,64] = feat[N,F] @ W[F,64] + b -------
// block = 128 threads (4 wave32's); wave w owns cols [16w,16w+16)
// grid.x = row tiles of 16; grid.y in 0..3 selects which (W,b,O) triple.
struct Mats { const float* W; const float* b; float* O; };

__global__ __launch_bounds__(128)
void gemm_qkvs_k(const float* __restrict__ feat, int F, int Nn,
                 Mats m0, Mats m1, Mats m2, Mats m3) {
    Mats m = (blockIdx.y == 0) ? m0 : (blockIdx.y == 1) ? m1
           : (blockIdx.y == 2) ? m2 : m3;
    const int lane    = threadIdx.x & 31;
    const int wave    = threadIdx.x >> 5;
    const int colbase = wave * 16;
    const int rowbase = blockIdx.x * 16;
    const int halfsel = lane >> 4;      // 0: K pair {0,1}; 1: K pair {2,3}
    const int l15     = lane & 15;
    int arow = rowbase + l15;
    if (arow >= Nn) arow = Nn - 1;      // clamp loads; stores are guarded
    const int koff = halfsel * 2;
    const float* frow = feat + (size_t)arow * F;

    v8f c = {};
    for (int k = 0; k < F; k += 4) {
        // A fragment: 16x4 fp32, row = l15, K = k+koff .. k+koff+1 (contiguous)
        const float2 av = *(const float2*)(frow + k + koff);
        v2f a; a[0] = av.x; a[1] = av.y;
        // B fragment: 4x16 fp32, col = colbase+l15, K rows mirror A's split
        v2f b;
        b[0] = m.W[(size_t)(k + koff)     * 64 + colbase + l15];
        b[1] = m.W[(size_t)(k + koff + 1) * 64 + colbase + l15];
        c = __builtin_amdgcn_wmma_f32_16x16x4_f32(
                false, a, false, b, (short)0, c, false, false);
    }
    const float bias = m.b[colbase + l15];
#pragma unroll
    for (int r = 0; r < 8; ++r) {
        const int row = rowbase + r + halfsel * 8;   // D: VGPR r -> M=r / M=r+8
        if (row < Nn) m.O[(size_t)row * 64 + colbase + l15] = c[r] + bias;
    }
}

// ---------------- edge pass 1: logits + segment max -------------------------
template <int H, int C>
__global__ void edge_logits_k(const float* __restrict__ q, const float* __restrict__ k,
                              const int* __restrict__ src, const int* __restrict__ dst,
                              float* __restrict__ alpha, unsigned* __restrict__ amax,
                              int E, float scale) {
    int t = blockIdx.x * blockDim.x + threadIdx.x;
    if (t >= E * H) return;
    const int e = t / H, h = t - e * H;
    const int s = src[e], d = dst[e];
    const float* qp = q + (size_t)d * 64 + h * C;
    const float* kp = k + (size_t)s * 64 + h * C;
    float dot = 0.0f;
#pragma unroll
    for (int c = 0; c < C; c += 4) {
        const float4 qv = *(const float4*)(qp + c);
        const float4 kv = *(const float4*)(kp + c);
        dot += qv.x * kv.x + qv.y * kv.y + qv.z * kv.z + qv.w * kv.w;
    }
    const float a = dot * scale;
    alpha[t] = a;
    atomicMax(amax + (size_t)d * H + h, fkey(a));
}

// ---------------- edge pass 2: softmax numerator + scatter accumulate -------
// PARTS splits the C channels across threads to spread atomic pressure.
template <int H, int C, int PARTS>
__global__ void edge_accum_k(const float* __restrict__ v,
                             const int* __restrict__ src, const int* __restrict__ dst,
                             const float* __restrict__ alpha, const unsigned* __restrict__ amax,
                             float* __restrict__ denom, float* __restrict__ accum, int E) {
    int t = blockIdx.x * blockDim.x + threadIdx.x;
    if (t >= E * H * PARTS) return;
    const int e   = t / (H * PARTS);
    const int rem = t - e * (H * PARTS);
    const int h   = rem / PARTS;
    const int p   = rem - h * PARTS;
    const int s = src[e], d = dst[e];
    float m = funkey(amax[(size_t)d * H + h]);
    if (!finitef(m)) m = 0.0f;
    const float ex = expf(alpha[(size_t)e * H + h] - m);
    if (p == 0) atomicAdd(denom + (size_t)d * H + h, ex);
    constexpr int CC = C / PARTS;
    const float* vp = v + (size_t)s * 64 + h * C + p * CC;
    float* ap = accum + (size_t)d * 64 + h * C + p * CC;
#pragma unroll
    for (int c = 0; c < CC; ++c) atomicAdd(ap + c, ex * vp[c]);
}

// ---------------- node finalize: normalize + skip (+ELU) --------------------
template <int H, int C, bool ELU>
__global__ void finalize_k(const float* __restrict__ accum, const float* __restrict__ denom,
                           const float* __restrict__ skip, float* __restrict__ out, int Nn) {
    int i = blockIdx.x * blockDim.x + threadIdx.x;
    if (i >= Nn * 64) return;
    const int n = i >> 6;
    const int j = i & 63;
    const int h = j / C;
    float val = accum[i] / (denom[(size_t)n * H + h] + 1e-16f) + skip[i];
    if (ELU) val = (val > 0.0f) ? val : (expf(val) - 1.0f);
    out[i] = val;
}

// ---------------- launch ----------------------------------------------------
extern "C" void kernel_launch(void* const* d_in, const int* in_sizes, int n_in,
                              void* d_out, int out_size, void* d_ws, size_t ws_size,
                              hipStream_t stream) {
    const float* x  = (const float*)d_in[0];
    const int*   ei = (const int*)d_in[1];
    const int N = in_sizes[0] / 128;
    const int E = in_sizes[1] / 2;
    const int* src = ei;
    const int* dst = ei + E;

    const float *wq0 = (const float*)d_in[2],  *bq0 = (const float*)d_in[3];
    const float *wk0 = (const float*)d_in[4],  *bk0 = (const float*)d_in[5];
    const float *wv0 = (const float*)d_in[6],  *bv0 = (const float*)d_in[7];
    const float *ws0 = (const float*)d_in[8],  *bs0 = (const float*)d_in[9];
    const float *wq1 = (const float*)d_in[10], *bq1 = (const float*)d_in[11];
    const float *wk1 = (const float*)d_in[12], *bk1 = (const float*)d_in[13];
    const float *wv1 = (const float*)d_in[14], *bv1 = (const float*)d_in[15];
    const float *ws1 = (const float*)d_in[16], *bs1 = (const float*)d_in[17];

    const size_t N64 = (size_t)N * 64;
    float*    q     = (float*)d_ws;
    float*    k     = q + N64;
    float*    v     = k + N64;
    float*    s     = v + N64;
    float*    hbuf  = s + N64;
    float*    accum = hbuf + N64;                 // [ accum | amax | denom ] zeroed together
    unsigned* amax  = (unsigned*)(accum + N64);
    float*    denom = (float*)amax + (size_t)N * 4;
    float*    alpha = denom + (size_t)N * 4;

    const int zn = N * 72;                        // accum(64) + amax(4) + denom(4) per node
    const dim3 gGemm((N + 15) / 16, 4);
    const int  bEdge = 256;

    // ---------------- layer 0: in=128, heads=4, ch=16, concat ---------------
    {
        Mats m0{wq0, bq0, q}, m1{wk0, bk0, k}, m2{wv0, bv0, v}, m3{ws0, bs0, s};
        gemm_qkvs_k<<<gGemm, 128, 0, stream>>>(x, 128, N, m0, m1, m2, m3);
    }
    fill_zero_k<<<(zn + 255) / 256, 256, 0, stream>>>(accum, zn);
    edge_logits_k<4, 16><<<(E * 4 + bEdge - 1) / bEdge, bEdge, 0, stream>>>(
        q, k, src, dst, alpha, amax, E, 0.25f);               // 1/sqrt(16)
    edge_accum_k<4, 16, 1><<<(E * 4 + bEdge - 1) / bEdge, bEdge, 0, stream>>>(
        v, src, dst, alpha, amax, denom, accum, E);
    finalize_k<4, 16, true><<<(N * 64 + 255) / 256, 256, 0, stream>>>(
        accum, denom, s, hbuf, N);

    // ---------------- layer 1: in=64, heads=1, ch=64, mean(=identity) -------
    {
        Mats m0{wq1, bq1, q}, m1{wk1, bk1, k}, m2{wv1, bv1, v}, m3{ws1, bs1, s};
        gemm_qkvs_k<<<gGemm, 128, 0, stream>>>(hbuf, 64, N, m0, m1, m2, m3);
    }
    fill_zero_k<<<(zn + 255) / 256, 256, 0, stream>>>(accum, zn);
    edge_logits_k<1, 64><<<(E + bEdge - 1) / bEdge, bEdge, 0, stream>>>(
        q, k, src, dst, alpha, amax, E, 0.125f);              // 1/sqrt(64)
    edge_accum_k<1, 64, 4><<<(E * 4 + bEdge - 1) / bEdge, bEdge, 0, stream>>>(
        v, src, dst, alpha, amax, denom, accum, E);
    finalize_k<1, 64, false><<<(N * 64 + 255) / 256, 256, 0, stream>>>(
        accum, denom, s, (float*)d_out, N);
}